// MPNEncoder_33148557590925
// MI455X (gfx1250) — compile-verified
//
#include <hip/hip_runtime.h>

// ---------------- problem constants (from reference) ----------------
#define NA     100000   // atoms
#define NBND   200000   // bonds
#define MAXNB  6
#define AF     133      // atom feature dim
#define BF     147      // bond feature dim
#define BFP    148      // bond feature dim padded to mult of 4
#define H      300      // hidden
#define HP     304      // hidden padded to 19*16
#define NM     5000     // molecules
#define NT     19       // HP / 16 (valid N tiles)
#define NTP    320      // padded B column count (5 groups * 4 tiles * 16)
#define NG     5        // N-tile groups of 4 per wave
#define H4     76       // HP / 4  (float4 per padded row)
#define KOUT   436      // AF + H = 433 padded to mult of 4

typedef __attribute__((ext_vector_type(2))) float v2f;
typedef __attribute__((ext_vector_type(8))) float v8f;

#define WMMA_F32(a, b, c) \
  __builtin_amdgcn_wmma_f32_16x16x4_f32(false, (a), false, (b), (short)0, (c), false, false)

// ---------------- fp32 WMMA GEMM ----------------
// One wave32 computes a 32x64 output slab (2 M-tiles x 4 N-tiles, 8
// accumulators) with V_WMMA_F32_16X16X4_F32. A is [M, lda] row-major
// zero-padded in K; BT is the weight matrix TRANSPOSED and zero-padded:
// BT[c * Kpad + k], c in [0,320). K-loop uses prefetch distance 2
// (triple buffering) with #pragma unroll 1: the fragments consumed each
// iteration were loaded two iterations earlier, so their loads have a
// full iteration of WMMA work to complete (partial s_wait_loadcnt).

__device__ __forceinline__ void wmma_epilogue(v8f acc, int m0, int half, int col,
                                              const float* __restrict__ Csrc,
                                              float bv,
                                              float* __restrict__ D,
                                              float* __restrict__ D2,
                                              int relu_d)
{
#pragma unroll
  for (int r = 0; r < 8; ++r) {
    const int row = m0 + r + (half << 3);     // C/D: VGPR r -> M = r + 8*half
    const size_t off = (size_t)row * HP + col;
    float x = acc[r] + bv;
    if (Csrc) x += Csrc[off];
    const float xr = fmaxf(x, 0.0f);
    D[off] = relu_d ? xr : x;
    if (D2) D2[off] = xr;
  }
}

__global__ void __launch_bounds__(128)
wmma_gemm_f32_kernel(const float* __restrict__ A, int lda,
                     const float* __restrict__ BT, int Kpad,
                     const float* __restrict__ Csrc,   // optional, ld = HP
                     const float* __restrict__ bias,   // optional, padded [320]
                     float* __restrict__ D,            // ld = HP
                     float* __restrict__ D2,           // optional, ld = HP
                     int M, int K, int relu_d)
{
  const int wid  = blockIdx.x * (blockDim.x >> 5) + (threadIdx.x >> 5);
  const int lane = threadIdx.x & 31;
  const int half = lane >> 4;     // K-half selector for A/B, M-half for C/D
  const int l15  = lane & 15;

  const int mslabs = M >> 5;      // M is an exact multiple of 32 here
  if (wid >= mslabs * NG) return; // uniform per wave -> EXEC stays all-ones
  const int ms = wid / NG;
  const int ng = wid - ms * NG;
  const int m0 = ms << 5;         // 32 rows per wave
  const int c0 = (ng << 6) + l15; // this lane's first of 4 column positions

  const float* __restrict__ ap0 = A  + (size_t)(m0 + l15)      * lda  + (half << 1);
  const float* __restrict__ ap1 = A  + (size_t)(m0 + 16 + l15) * lda  + (half << 1);
  const float* __restrict__ bp0 = BT + (size_t)(c0)            * Kpad + (half << 1);
  const float* __restrict__ bp1 = BT + (size_t)(c0 + 16)       * Kpad + (half << 1);
  const float* __restrict__ bp2 = BT + (size_t)(c0 + 32)       * Kpad + (half << 1);
  const float* __restrict__ bp3 = BT + (size_t)(c0 + 48)       * Kpad + (half << 1);

  v8f acc00 = {0.f,0.f,0.f,0.f,0.f,0.f,0.f,0.f};
  v8f acc01 = {0.f,0.f,0.f,0.f,0.f,0.f,0.f,0.f};
  v8f acc02 = {0.f,0.f,0.f,0.f,0.f,0.f,0.f,0.f};
  v8f acc03 = {0.f,0.f,0.f,0.f,0.f,0.f,0.f,0.f};
  v8f acc10 = {0.f,0.f,0.f,0.f,0.f,0.f,0.f,0.f};
  v8f acc11 = {0.f,0.f,0.f,0.f,0.f,0.f,0.f,0.f};
  v8f acc12 = {0.f,0.f,0.f,0.f,0.f,0.f,0.f,0.f};
  v8f acc13 = {0.f,0.f,0.f,0.f,0.f,0.f,0.f,0.f};

  // --- triple-buffered K loop (K >= 8, multiple of 4) ---
  v2f a0_0 = *(const v2f*)(ap0);       // set 0: k-step 0
  v2f a1_0 = *(const v2f*)(ap1);
  v2f b0_0 = *(const v2f*)(bp0);
  v2f b1_0 = *(const v2f*)(bp1);
  v2f b2_0 = *(const v2f*)(bp2);
  v2f b3_0 = *(const v2f*)(bp3);
  v2f a0_1 = *(const v2f*)(ap0 + 4);   // set 1: k-step 1
  v2f a1_1 = *(const v2f*)(ap1 + 4);
  v2f b0_1 = *(const v2f*)(bp0 + 4);
  v2f b1_1 = *(const v2f*)(bp1 + 4);
  v2f b2_1 = *(const v2f*)(bp2 + 4);
  v2f b3_1 = *(const v2f*)(bp3 + 4);

#pragma unroll 1
  for (int kk = 8; kk < K; kk += 4) {
    // prefetch set 2 (distance 2)
    const v2f a0_2 = *(const v2f*)(ap0 + kk);
    const v2f a1_2 = *(const v2f*)(ap1 + kk);
    const v2f b0_2 = *(const v2f*)(bp0 + kk);
    const v2f b1_2 = *(const v2f*)(bp1 + kk);
    const v2f b2_2 = *(const v2f*)(bp2 + kk);
    const v2f b3_2 = *(const v2f*)(bp3 + kk);
    // consume set 0 (loaded two iterations ago)
    acc00 = WMMA_F32(a0_0, b0_0, acc00);
    acc10 = WMMA_F32(a1_0, b0_0, acc10);
    acc01 = WMMA_F32(a0_0, b1_0, acc01);
    acc11 = WMMA_F32(a1_0, b1_0, acc11);
    acc02 = WMMA_F32(a0_0, b2_0, acc02);
    acc12 = WMMA_F32(a1_0, b2_0, acc12);
    acc03 = WMMA_F32(a0_0, b3_0, acc03);
    acc13 = WMMA_F32(a1_0, b3_0, acc13);
    // rotate
    a0_0 = a0_1; a1_0 = a1_1;
    b0_0 = b0_1; b1_0 = b1_1; b2_0 = b2_1; b3_0 = b3_1;
    a0_1 = a0_2; a1_1 = a1_2;
    b0_1 = b0_2; b1_1 = b1_2; b2_1 = b2_2; b3_1 = b3_2;
  }
  // drain: two remaining staged sets
  acc00 = WMMA_F32(a0_0, b0_0, acc00);
  acc10 = WMMA_F32(a1_0, b0_0, acc10);
  acc01 = WMMA_F32(a0_0, b1_0, acc01);
  acc11 = WMMA_F32(a1_0, b1_0, acc11);
  acc02 = WMMA_F32(a0_0, b2_0, acc02);
  acc12 = WMMA_F32(a1_0, b2_0, acc12);
  acc03 = WMMA_F32(a0_0, b3_0, acc03);
  acc13 = WMMA_F32(a1_0, b3_0, acc13);
  acc00 = WMMA_F32(a0_1, b0_1, acc00);
  acc10 = WMMA_F32(a1_1, b0_1, acc10);
  acc01 = WMMA_F32(a0_1, b1_1, acc01);
  acc11 = WMMA_F32(a1_1, b1_1, acc11);
  acc02 = WMMA_F32(a0_1, b2_1, acc02);
  acc12 = WMMA_F32(a1_1, b2_1, acc12);
  acc03 = WMMA_F32(a0_1, b3_1, acc03);
  acc13 = WMMA_F32(a1_1, b3_1, acc13);

  const int ntb = ng << 2;   // first N-tile index of this group
  const float bv0 = bias ? bias[c0]      : 0.0f;
  const float bv1 = bias ? bias[c0 + 16] : 0.0f;
  const float bv2 = bias ? bias[c0 + 32] : 0.0f;
  const float bv3 = bias ? bias[c0 + 48] : 0.0f;
  if (ntb + 0 < NT) {
    wmma_epilogue(acc00, m0,      half, c0,      Csrc, bv0, D, D2, relu_d);
    wmma_epilogue(acc10, m0 + 16, half, c0,      Csrc, bv0, D, D2, relu_d);
  }
  if (ntb + 1 < NT) {
    wmma_epilogue(acc01, m0,      half, c0 + 16, Csrc, bv1, D, D2, relu_d);
    wmma_epilogue(acc11, m0 + 16, half, c0 + 16, Csrc, bv1, D, D2, relu_d);
  }
  if (ntb + 2 < NT) {
    wmma_epilogue(acc02, m0,      half, c0 + 32, Csrc, bv2, D, D2, relu_d);
    wmma_epilogue(acc12, m0 + 16, half, c0 + 32, Csrc, bv2, D, D2, relu_d);
  }
  if (ntb + 3 < NT) {
    wmma_epilogue(acc03, m0,      half, c0 + 48, Csrc, bv3, D, D2, relu_d);
    wmma_epilogue(acc13, m0 + 16, half, c0 + 48, Csrc, bv3, D, D2, relu_d);
  }
}

// ---------------- prep kernels (tiny / one-shot) ----------------

__global__ void zero_kernel(float* __restrict__ p, int n) {
  for (int t = blockIdx.x * blockDim.x + threadIdx.x; t < n;
       t += gridDim.x * blockDim.x)
    p[t] = 0.0f;
}

// BT[c * Kpad + k] = (k < K && c < N) ? W[k * N + c] : 0 ; c in [0,320)
__global__ void transpose_pad_kernel(const float* __restrict__ W, int K, int N,
                                     int Kpad, float* __restrict__ BT)
{
  int t = blockIdx.x * blockDim.x + threadIdx.x;
  if (t >= NTP * Kpad) return;
  const int c = t / Kpad;
  const int k = t - c * Kpad;
  BT[t] = (k < K && c < N) ? W[(size_t)k * N + c] : 0.0f;
}

__global__ void pad_bias_kernel(const float* __restrict__ b, float* __restrict__ bp) {
  int t = blockIdx.x * blockDim.x + threadIdx.x;
  if (t < NTP) bp[t] = (t < H) ? b[t] : 0.0f;
}

// fb_p [NBND, 148] zero-padded copy of f_bonds [NBND, 147]
__global__ void pad_bonds_kernel(const float* __restrict__ fb, float* __restrict__ fbp) {
  int t = blockIdx.x * blockDim.x + threadIdx.x;
  if (t >= NBND * BFP) return;
  const int b = t / BFP;
  const int k = t - b * BFP;
  fbp[t] = (k < BF) ? fb[(size_t)b * BF + k] : 0.0f;
}

// a_input [NA, 436] = [f_atoms | a_message | 0-pad]
__global__ void concat_kernel(const float* __restrict__ f_atoms,
                              const float* __restrict__ a_message,
                              float* __restrict__ a_input)
{
  int t = blockIdx.x * blockDim.x + threadIdx.x;
  if (t >= NA * KOUT) return;
  const int a = t / KOUT;
  const int k = t - a * KOUT;
  float v;
  if (k < AF)            v = f_atoms[(size_t)a * AF + k];
  else if (k < AF + H)   v = a_message[(size_t)a * HP + (k - AF)];
  else                   v = 0.0f;
  a_input[t] = v;
}

// ---------------- memory-bound glue kernels (float4) ----------------

// a_message[a,:] = sum_j message[a2b[a][j],:]
__global__ void aggregate_kernel(const float* __restrict__ message,
                                 const int* __restrict__ a2b,
                                 float* __restrict__ a_message)
{
  int t = blockIdx.x * blockDim.x + threadIdx.x;
  if (t >= NA * H4) return;
  const int a = t / H4;
  const int c = (t - a * H4) << 2;
  const int* nb = a2b + (size_t)a * MAXNB;
  float4 s = make_float4(0.f, 0.f, 0.f, 0.f);
#pragma unroll
  for (int j = 0; j < MAXNB; ++j) {
    const float4 v = *(const float4*)(message + (size_t)nb[j] * HP + c);
    s.x += v.x; s.y += v.y; s.z += v.z; s.w += v.w;
  }
  *(float4*)(a_message + (size_t)a * HP + c) = s;
}

// m_in[b,:] = a_message[b2a[b],:] - message[b2revb[b],:]
__global__ void bond_combine_kernel(const float* __restrict__ a_message,
                                    const float* __restrict__ message,
                                    const int* __restrict__ b2a,
                                    const int* __restrict__ b2revb,
                                    float* __restrict__ m_in)
{
  int t = blockIdx.x * blockDim.x + threadIdx.x;
  if (t >= NBND * H4) return;
  const int b = t / H4;
  const int c = (t - b * H4) << 2;
  const float4 am = *(const float4*)(a_message + (size_t)b2a[b]    * HP + c);
  const float4 rm = *(const float4*)(message   + (size_t)b2revb[b] * HP + c);
  float4 o;
  o.x = am.x - rm.x; o.y = am.y - rm.y; o.z = am.z - rm.z; o.w = am.w - rm.w;
  *(float4*)(m_in + (size_t)b * HP + c) = o;
}

__global__ void counts_kernel(const int* __restrict__ a2mol,
                              float* __restrict__ counts)
{
  int t = blockIdx.x * blockDim.x + threadIdx.x;
  if (t >= NA) return;
  atomicAdd(&counts[a2mol[t]], 1.0f);
}

__global__ void pool_kernel(const float* __restrict__ ah,
                            const int* __restrict__ a2mol,
                            float* __restrict__ out)
{
  int t = blockIdx.x * blockDim.x + threadIdx.x;
  if (t >= NA * H) return;
  const int a = t / H;
  const int h = t - a * H;
  atomicAdd(&out[(size_t)a2mol[a] * H + h], ah[(size_t)a * HP + h]);
}

__global__ void finalize_kernel(float* __restrict__ out,
                                const float* __restrict__ counts)
{
  int t = blockIdx.x * blockDim.x + threadIdx.x;
  if (t >= NM * H) return;
  out[t] /= fmaxf(counts[t / H], 1.0f);
}

// ---------------- launcher ----------------
extern "C" void kernel_launch(void* const* d_in, const int* in_sizes, int n_in,
                              void* d_out, int out_size, void* d_ws, size_t ws_size,
                              hipStream_t stream) {
  const float* f_atoms = (const float*)d_in[0];
  const float* f_bonds = (const float*)d_in[1];
  const int*   a2b     = (const int*)  d_in[2];
  const int*   b2a     = (const int*)  d_in[3];
  const int*   b2revb  = (const int*)  d_in[4];
  const int*   a2mol   = (const int*)  d_in[5];
  const float* W_i     = (const float*)d_in[7];
  const float* W_h     = (const float*)d_in[8];
  const float* W_o     = (const float*)d_in[9];
  const float* b_o     = (const float*)d_in[10];
  float* out = (float*)d_out;

  // workspace layout (regions reused across disjoint lifetimes)
  char* ws = (char*)d_ws;
  const size_t SZ_B = (size_t)NBND * HP * sizeof(float);   // 243.2 MB
  const size_t SZ_A = (size_t)NA   * HP * sizeof(float);   // 121.6 MB
  float* inp          = (float*)(ws);                 // r0
  float* message      = (float*)(ws + SZ_B);          // r1
  float* r2           = (float*)(ws + 2 * SZ_B);      // fb_pad -> m_in -> a_input
  float* a_message    = (float*)(ws + 3 * SZ_B);      // r3
  float* fb_pad       = r2;
  float* m_in         = r2;
  float* a_input      = r2;
  float* atom_hiddens = inp;                          // inp dead after depth loop
  char*  small = ws + 3 * SZ_B + SZ_A;
  float* counts = (float*)(small);               small += (size_t)NM * 4;
  float* WiT    = (float*)(small);               small += (size_t)NTP * BFP * 4;
  float* WhT    = (float*)(small);               small += (size_t)NTP * H * 4;
  float* WoT    = (float*)(small);               small += (size_t)NTP * KOUT * 4;
  float* bias_p = (float*)(small);

  const int TPB = 256;
  const int GEMM_TPB = 128;                  // 4 wave32 per block
  const int WPB = GEMM_TPB / 32;
  const int gB = ((NBND / 32) * NG + WPB - 1) / WPB;
  const int gA = ((NA   / 32) * NG + WPB - 1) / WPB;

  // ---- prep ----
  zero_kernel<<<(NM * H + TPB - 1) / TPB, TPB, 0, stream>>>(out, NM * H);
  zero_kernel<<<(NM + TPB - 1) / TPB, TPB, 0, stream>>>(counts, NM);
  transpose_pad_kernel<<<(NTP * BFP  + TPB - 1) / TPB, TPB, 0, stream>>>(W_i, BF,     H, BFP,  WiT);
  transpose_pad_kernel<<<(NTP * H    + TPB - 1) / TPB, TPB, 0, stream>>>(W_h, H,      H, H,    WhT);
  transpose_pad_kernel<<<(NTP * KOUT + TPB - 1) / TPB, TPB, 0, stream>>>(W_o, AF + H, H, KOUT, WoT);
  pad_bias_kernel<<<(NTP + TPB - 1) / TPB, TPB, 0, stream>>>(b_o, bias_p);
  pad_bonds_kernel<<<(NBND * BFP + TPB - 1) / TPB, TPB, 0, stream>>>(f_bonds, fb_pad);

  // ---- inp = f_bonds @ W_i ; message = relu(inp) ----
  wmma_gemm_f32_kernel<<<gB, GEMM_TPB, 0, stream>>>(
      fb_pad, BFP, WiT, BFP, nullptr, nullptr,
      inp, message, NBND, BFP, /*relu_d=*/0);

  // ---- DEPTH-1 = 2 message-passing iterations ----
  for (int d = 0; d < 2; ++d) {
    aggregate_kernel<<<(NA * H4 + TPB - 1) / TPB, TPB, 0, stream>>>(
        message, a2b, a_message);
    bond_combine_kernel<<<(NBND * H4 + TPB - 1) / TPB, TPB, 0, stream>>>(
        a_message, message, b2a, b2revb, m_in);
    // message = relu(inp + m_in @ W_h)
    wmma_gemm_f32_kernel<<<gB, GEMM_TPB, 0, stream>>>(
        m_in, HP, WhT, H, inp, nullptr,
        message, nullptr, NBND, H, /*relu_d=*/1);
  }

  // ---- output layer on concat([f_atoms, a_message]) ----
  aggregate_kernel<<<(NA * H4 + TPB - 1) / TPB, TPB, 0, stream>>>(
      message, a2b, a_message);
  concat_kernel<<<(NA * KOUT + TPB - 1) / TPB, TPB, 0, stream>>>(
      f_atoms, a_message, a_input);
  wmma_gemm_f32_kernel<<<gA, GEMM_TPB, 0, stream>>>(
      a_input, KOUT, WoT, KOUT, nullptr, bias_p,
      atom_hiddens, nullptr, NA, KOUT, /*relu_d=*/1);

  // ---- segment mean over molecules ----
  counts_kernel<<<(NA + TPB - 1) / TPB, TPB, 0, stream>>>(a2mol, counts);
  pool_kernel<<<(NA * H + TPB - 1) / TPB, TPB, 0, stream>>>(
      atom_hiddens, a2mol, out);
  finalize_kernel<<<(NM * H + TPB - 1) / TPB, TPB, 0, stream>>>(out, counts);
}